// ChildSumTreeLSTM_8847632630373
// MI455X (gfx1250) — compile-verified
//
#include <hip/hip_runtime.h>
#include <math.h>

typedef __attribute__((ext_vector_type(2))) float v2f;
typedef __attribute__((ext_vector_type(8))) float v8f;

#define IN_DIM 300
#define MEM    256
#define DEPTH  15
#define NNODES ((1 << (DEPTH + 1)) - 1)

// ---- dynamic LDS layout (in floats) ----
#define OFF_X   0                       // 16 x 300
#define OFF_HL  (OFF_X + 16 * IN_DIM)   // 16 x 256
#define OFF_HR  (OFF_HL + 16 * MEM)
#define OFF_CL  (OFF_HR + 16 * MEM)
#define OFF_CR  (OFF_CL + 16 * MEM)
#define OFF_FX  (OFF_CR + 16 * MEM)     // fx preact + b_fx
#define OFF_FC  (OFF_FX + 16 * MEM)     // fl*cl + fr*cr
#define OFF_IOU (OFF_FC + 16 * MEM)     // 16 x 768 iou preact
#define LDS_FLOATS (OFF_IOU + 16 * 3 * MEM)
#define LDS_BYTES  (LDS_FLOATS * sizeof(float))

__device__ __forceinline__ float sigmoidf_(float x) {
  return 1.0f / (1.0f + expf(-x));
}

// acc += A(16 x K, from LDS at aoff with row stride astride) * B(K x 16),
// where B(k, n) = W[colbase + n][k], W row-major with row stride wstride.
// A lane mapping: m = lane&15, K-pair offset kh = 2*(lane>>4).
__device__ __forceinline__ v8f wmma_acc_lds_w(
    const float* smem, int aoff, int astride,
    const float* __restrict__ W, int wstride, int colbase,
    int K, int lane, v8f acc)
{
  const int m  = lane & 15;
  const int kh = (lane >> 4) << 1;      // 0 or 2
  const float* __restrict__ wrow = W + (size_t)(colbase + m) * wstride;
  const float* arow = smem + aoff + m * astride;
  for (int k = 0; k < K; k += 4) {
    v2f a, b;
    a.x = arow[k + kh];
    a.y = arow[k + kh + 1];
    b.x = wrow[k + kh];
    b.y = wrow[k + kh + 1];
    acc = __builtin_amdgcn_wmma_f32_16x16x4_f32(
        /*neg_a=*/false, a, /*neg_b=*/false, b,
        /*c_mod=*/(short)0, acc, /*reuse_a=*/false, /*reuse_b=*/false);
  }
  return acc;
}

// Same, but A(m,k) = hl(m,k) + hr(m,k) computed on the fly (child_h_sum), K = MEM.
__device__ __forceinline__ v8f wmma_acc_hsum_w(
    const float* smem, const float* __restrict__ W, int colbase, int lane, v8f acc)
{
  const int m  = lane & 15;
  const int kh = (lane >> 4) << 1;
  const float* __restrict__ wrow = W + (size_t)(colbase + m) * MEM;
  for (int k = 0; k < MEM; k += 4) {
    const int ia = m * MEM + k + kh;
    v2f a, b;
    a.x = smem[OFF_HL + ia]     + smem[OFF_HR + ia];
    a.y = smem[OFF_HL + ia + 1] + smem[OFF_HR + ia + 1];
    b.x = wrow[k + kh];
    b.y = wrow[k + kh + 1];
    acc = __builtin_amdgcn_wmma_f32_16x16x4_f32(
        false, a, false, b, (short)0, acc, false, false);
  }
  return acc;
}

__global__ void treelstm_level_kernel(
    const float* __restrict__ x_all,    // [N, 300]
    const float* __restrict__ W_ioux,   // [768, 300]
    const float* __restrict__ b_ioux,   // [768]
    const float* __restrict__ W_iouh,   // [768, 256]
    const float* __restrict__ b_iouh,   // [768]
    const float* __restrict__ W_fx,     // [256, 300]
    const float* __restrict__ b_fx,     // [256]
    const float* __restrict__ W_fh,     // [256, 256]
    const float* __restrict__ b_fh,     // [256]
    float* __restrict__ c_out,          // [N, 256]
    float* __restrict__ h_out,          // [N, 256]
    int start, int size, int leaf)
{
  extern __shared__ float smem[];
  const int tid  = threadIdx.x;
  const int lane = tid & 31;
  const int wave = tid >> 5;
  const int row0 = blockIdx.x * 16;

  // ---------- gather: x rows, children h/c ----------
  for (int e = tid; e < 16 * IN_DIM; e += 256) {
    int m = e / IN_DIM, k = e - m * IN_DIM;
    int rm = row0 + m; if (rm >= size) rm = size - 1;
    smem[OFF_X + m * IN_DIM + k] = x_all[(size_t)(start + rm) * IN_DIM + k];
  }
  if (!leaf) {
    const int cstart = 2 * start + 1;   // base of the child level
    for (int e = tid; e < 16 * MEM; e += 256) {
      int m = e >> 8, k = e & (MEM - 1);
      int rm = row0 + m; if (rm >= size) rm = size - 1;
      size_t li = (size_t)(cstart + 2 * rm) * MEM + k;
      size_t ri = li + MEM;             // right child is next node
      smem[OFF_HL + e] = h_out[li];
      smem[OFF_HR + e] = h_out[ri];
      smem[OFF_CL + e] = c_out[li];
      smem[OFF_CR + e] = c_out[ri];
    }
  }
  __syncthreads();

  if (!leaf) {
    // ---------- fx = x @ W_fx^T + b_fx ----------
    for (int t = wave; t < 16; t += 8) {
      const int colbase = t * 16;
      v8f acc = {};
      acc = wmma_acc_lds_w(smem, OFF_X, IN_DIM, W_fx, IN_DIM, colbase, IN_DIM, lane, acc);
      const int n = colbase + (lane & 15);
      const float bias = b_fx[n];
      const int mh = (lane >> 4) << 3;
      #pragma unroll
      for (int r = 0; r < 8; ++r)
        smem[OFF_FX + (mh + r) * MEM + n] = acc[r] + bias;
    }
    __syncthreads();
    // ---------- fl/fr gates, fc = fl*cl + fr*cr ----------
    for (int t = wave; t < 16; t += 8) {
      const int colbase = t * 16;
      v8f gl = {}, gr = {};
      gl = wmma_acc_lds_w(smem, OFF_HL, MEM, W_fh, MEM, colbase, MEM, lane, gl);
      gr = wmma_acc_lds_w(smem, OFF_HR, MEM, W_fh, MEM, colbase, MEM, lane, gr);
      const int n = colbase + (lane & 15);
      const float bias = b_fh[n];
      const int mh = (lane >> 4) << 3;
      #pragma unroll
      for (int r = 0; r < 8; ++r) {
        const int m = mh + r;
        const float fxv = smem[OFF_FX + m * MEM + n];
        const float fl = sigmoidf_(gl[r] + bias + fxv);
        const float fr = sigmoidf_(gr[r] + bias + fxv);
        smem[OFF_FC + m * MEM + n] =
            fl * smem[OFF_CL + m * MEM + n] + fr * smem[OFF_CR + m * MEM + n];
      }
    }
  } else {
    for (int e = tid; e < 16 * MEM; e += 256) smem[OFF_FC + e] = 0.0f;
  }
  __syncthreads();

  // ---------- iou = x @ W_ioux^T + hsum @ W_iouh^T + b_ioux + b_iouh ----------
  for (int t = wave; t < 48; t += 8) {
    const int colbase = t * 16;
    v8f acc = {};
    acc = wmma_acc_lds_w(smem, OFF_X, IN_DIM, W_ioux, IN_DIM, colbase, IN_DIM, lane, acc);
    if (!leaf)
      acc = wmma_acc_hsum_w(smem, W_iouh, colbase, lane, acc);
    const int n = colbase + (lane & 15);
    const float bias = b_ioux[n] + b_iouh[n];
    const int mh = (lane >> 4) << 3;
    #pragma unroll
    for (int r = 0; r < 8; ++r)
      smem[OFF_IOU + (mh + r) * (3 * MEM) + n] = acc[r] + bias;
  }
  __syncthreads();

  // ---------- gates + writeback ----------
  for (int e = tid; e < 16 * MEM; e += 256) {
    const int m = e >> 8, n = e & (MEM - 1);
    const int rm = row0 + m;
    if (rm < size) {
      const float iv = smem[OFF_IOU + m * (3 * MEM) + n];
      const float ov = smem[OFF_IOU + m * (3 * MEM) + MEM + n];
      const float uv = smem[OFF_IOU + m * (3 * MEM) + 2 * MEM + n];
      const float cn = sigmoidf_(iv) * tanhf(uv) + smem[OFF_FC + e];
      const float hn = sigmoidf_(ov) * tanhf(cn);
      const size_t o = (size_t)(start + rm) * MEM + n;
      c_out[o] = cn;
      h_out[o] = hn;
    }
  }
}

extern "C" void kernel_launch(void* const* d_in, const int* in_sizes, int n_in,
                              void* d_out, int out_size, void* d_ws, size_t ws_size,
                              hipStream_t stream) {
  (void)in_sizes; (void)n_in; (void)out_size; (void)d_ws; (void)ws_size;
  const float* x      = (const float*)d_in[0];
  const float* W_ioux = (const float*)d_in[1];
  const float* b_ioux = (const float*)d_in[2];
  const float* W_iouh = (const float*)d_in[3];
  const float* b_iouh = (const float*)d_in[4];
  const float* W_fx   = (const float*)d_in[5];
  const float* b_fx   = (const float*)d_in[6];
  const float* W_fh   = (const float*)d_in[7];
  const float* b_fh   = (const float*)d_in[8];

  float* c_out = (float*)d_out;                     // [N, MEM]
  float* h_out = c_out + (size_t)NNODES * MEM;      // [N, MEM]

  for (int d = DEPTH; d >= 0; --d) {
    const int start = (1 << d) - 1;
    const int size  = 1 << d;
    const int tiles = (size + 15) / 16;
    treelstm_level_kernel<<<tiles, 256, LDS_BYTES, stream>>>(
        x, W_ioux, b_ioux, W_iouh, b_iouh, W_fx, b_fx, W_fh, b_fh,
        c_out, h_out, start, size, (d == DEPTH) ? 1 : 0);
  }
}